// TokenAlignHybridLoss_86105504350470
// MI455X (gfx1250) — compile-verified
//
#include <hip/hip_runtime.h>
#include <math.h>

#define BDIM 8192
#define MARGIN 0.2f
#define LAMBDA 0.1f
#define EPS 1e-8f
#define COLBLK 64

typedef float v2f __attribute__((ext_vector_type(2)));
typedef float v8f __attribute__((ext_vector_type(8)));

__device__ __forceinline__ float vsel8(v8f d, int v) {
    switch (v) {
        case 0: return d[0]; case 1: return d[1]; case 2: return d[2]; case 3: return d[3];
        case 4: return d[4]; case 5: return d[5]; case 6: return d[6]; default: return d[7];
    }
}

// ---------------------------------------------------------------------------
// Row pass: one wave per 16-row tile. Computes
//   rdot[i] = dot(S[i,:], S[ha[i],:])   via diag of WMMA-accumulated 16x16 gram
//   rn2[i]  = ||S[i,:]||^2              via fused per-lane FMA + shfl_xor(16)
// A-matrix 16x4 f32 layout: lane&15 = M, lane>>4 selects K-half {0,1}|{2,3}
// B-matrix  4x16 f32 layout: lane&15 = N, lane>>4 selects K-half  (mirrors A)
// ---------------------------------------------------------------------------
__global__ void __launch_bounds__(256) row_pass(const float* __restrict__ S,
                                                const int* __restrict__ ha,
                                                float* __restrict__ rdot,
                                                float* __restrict__ rn2) {
    const int lane = threadIdx.x & 31;
    const int wave = threadIdx.x >> 5;
    const int tile = blockIdx.x * (blockDim.x >> 5) + wave;
    const int i0 = tile * 16;
    const int m  = lane & 15;
    const int kh = lane >> 4;           // 0 -> K={0,1}, 1 -> K={2,3}

    const float* Ap = S + (size_t)(i0 + m) * BDIM + 2 * kh;        // my row
    const int    gr = ha[i0 + m];
    const float* Bp = S + (size_t)gr * BDIM + 2 * kh;              // gathered row

    v8f acc = {};
    float sq = 0.f;
    for (int k0 = 0; k0 < BDIM; k0 += 4) {
        v2f a = *(const v2f*)(Ap + k0);
        v2f b = *(const v2f*)(Bp + k0);
        sq = fmaf(a.x, a.x, sq);
        sq = fmaf(a.y, a.y, sq);
        // D = A(16x4) * B(4x16) + C : accumulate full gram; diagonal is what we need
        acc = __builtin_amdgcn_wmma_f32_16x16x4_f32(false, a, false, b,
                                                    (short)0, acc, false, false);
    }

    // rn2: lane l (K-half 0) + lane l+16 (K-half 1) hold complementary partials
    sq += __shfl_xor(sq, 16, 32);
    if (lane < 16) rn2[i0 + lane] = sq;

    // diag extraction from C/D layout:
    //   lanes 0-15 : VGPR v holds (M=v,   N=lane)    -> diag (v,v)   at lane v,    v<8
    //   lanes 16-31: VGPR v holds (M=v+8, N=lane-16) -> diag (m,m)   at lane m+16, VGPR m-8
    if (lane < 8) {
        rdot[i0 + lane] = vsel8(acc, lane);
    } else if (lane >= 24) {
        rdot[i0 + lane - 16] = vsel8(acc, lane - 24);
    }
}

// ---------------------------------------------------------------------------
// Column sum-of-squares: cn2[j] = sum_r S[r,j]^2 ; fully coalesced, register acc
// ---------------------------------------------------------------------------
__global__ void __launch_bounds__(256) col_sumsq(const float* __restrict__ S,
                                                 float* __restrict__ cn2) {
    const int j = blockIdx.x * blockDim.x + threadIdx.x;
    float acc = 0.f;
    for (int r = 0; r < BDIM; ++r) {
        float v = S[(size_t)r * BDIM + j];
        acc = fmaf(v, v, acc);
    }
    cn2[j] = acc;
}

// ---------------------------------------------------------------------------
// Column-dot partial pass: block b covers rows [b*128, b*128+128).
// Row staged in LDS (coalesced), partial pc[i] += row[i]*row[ht[i]] with each
// i owned by one fixed thread (deterministic, race-free). Partials -> ws.
// ---------------------------------------------------------------------------
__global__ void __launch_bounds__(256) col_pass(const float* __restrict__ S,
                                                const int* __restrict__ ht,
                                                float* __restrict__ pcd) {
    __shared__ float rowbuf[BDIM];   // 32 KB
    __shared__ float pc[BDIM];       // 32 KB
    const int tid = threadIdx.x;
    const int nt  = blockDim.x;

    for (int i = tid; i < BDIM; i += nt) pc[i] = 0.f;
    __syncthreads();

    const int r0 = blockIdx.x * (BDIM / COLBLK);
    const int r1 = r0 + (BDIM / COLBLK);
    for (int r = r0; r < r1; ++r) {
        for (int j = tid; j < BDIM; j += nt)
            rowbuf[j] = S[(size_t)r * BDIM + j];
        __syncthreads();
        for (int i = tid; i < BDIM; i += nt)
            pc[i] = fmaf(rowbuf[i], rowbuf[ht[i]], pc[i]);
        __syncthreads();
    }

    for (int i = tid; i < BDIM; i += nt)
        pcd[(size_t)blockIdx.x * BDIM + i] = pc[i];
}

__global__ void __launch_bounds__(256) col_reduce(const float* __restrict__ pcd,
                                                  float* __restrict__ cdot) {
    const int i = blockIdx.x * blockDim.x + threadIdx.x;
    float a = 0.f;
    for (int b = 0; b < COLBLK; ++b) a += pcd[(size_t)b * BDIM + i];
    cdot[i] = a;
}

// ---------------------------------------------------------------------------
// Final: triplet gathers + cosine assembly + masked mean reduction -> 3 scalars
// ---------------------------------------------------------------------------
__global__ void __launch_bounds__(256) final_reduce(const float* __restrict__ S,
                                                    const unsigned char* __restrict__ pm,
                                                    const int* __restrict__ ha,
                                                    const int* __restrict__ ht,
                                                    const float* __restrict__ rdot,
                                                    const float* __restrict__ rn2,
                                                    const float* __restrict__ cdot,
                                                    const float* __restrict__ cn2,
                                                    float* __restrict__ out) {
    __shared__ float st[256], sc[256], sn[256];
    const int tid = threadIdx.x;
    float at = 0.f, ac = 0.f, an = 0.f;

    for (int i = tid; i < BDIM; i += 256) {
        const float mk  = pm[(size_t)i * BDIM + i] ? 1.f : 0.f;
        const int   hai = ha[i];
        const int   hti = ht[i];
        const float pos  = S[(size_t)i * BDIM + i];
        const float tneg = S[(size_t)i * BDIM + hti];
        const float aneg = S[(size_t)hai * BDIM + i];
        const float trip = fmaxf(MARGIN - pos + tneg, 0.f)
                         + fmaxf(MARGIN - pos + aneg, 0.f);
        const float cosa = rdot[i] / fmaxf(sqrtf(rn2[i]) * sqrtf(rn2[hai]), EPS);
        const float cost = cdot[i] / fmaxf(sqrtf(cn2[i]) * sqrtf(cn2[hti]), EPS);
        const float cons = fabsf(cosa - cost);   // relu(|x| - 0) == |x|
        at += trip * mk;
        ac += cons * mk;
        an += mk;
    }

    st[tid] = at; sc[tid] = ac; sn[tid] = an;
    __syncthreads();
    for (int s = 128; s > 0; s >>= 1) {
        if (tid < s) { st[tid] += st[tid + s]; sc[tid] += sc[tid + s]; sn[tid] += sn[tid + s]; }
        __syncthreads();
    }
    if (tid == 0) {
        const float cnt = fmaxf(sn[0], 1.f);
        const float tl = st[0] / cnt;
        const float cl = sc[0] / cnt;
        out[0] = tl + LAMBDA * cl;
        out[1] = tl;
        out[2] = cl;
    }
}

extern "C" void kernel_launch(void* const* d_in, const int* in_sizes, int n_in,
                              void* d_out, int out_size, void* d_ws, size_t ws_size,
                              hipStream_t stream) {
    const float*         S  = (const float*)d_in[0];
    const unsigned char* pm = (const unsigned char*)d_in[1];
    const int*           ha = (const int*)d_in[2];
    const int*           ht = (const int*)d_in[3];
    float* out = (float*)d_out;

    float* w    = (float*)d_ws;
    float* rdot = w;                  // B
    float* rn2  = w + BDIM;           // B
    float* cdot = w + 2 * BDIM;       // B
    float* cn2  = w + 3 * BDIM;       // B
    float* pcd  = w + 4 * BDIM;       // COLBLK * B partials

    // 512 tiles of 16 rows, 8 waves (tiles) per 256-thread block -> 64 blocks
    row_pass<<<(BDIM / 16) / 8, 256, 0, stream>>>(S, ha, rdot, rn2);
    col_sumsq<<<BDIM / 256, 256, 0, stream>>>(S, cn2);
    col_pass<<<COLBLK, 256, 0, stream>>>(S, ht, pcd);
    col_reduce<<<BDIM / 256, 256, 0, stream>>>(pcd, cdot);
    final_reduce<<<1, 256, 0, stream>>>(S, pm, ha, ht, rdot, rn2, cdot, cn2, out);
}